// HexPlaneField_36910948941842
// MI455X (gfx1250) — compile-verified
//
#include <hip/hip_runtime.h>
#include <cstdint>

// HexPlane: RESOLUTION=(128,128,128,150), F=16, MULTIRES=(1,2),
// PAIRS=((0,1),(0,2),(1,2),(0,3),(1,3),(2,3))
// Plane k has shape (F=16, H=res[i], W=res[j]); gx=c[i] indexes W, gy=c[j] indexes H.
constexpr int HH[12] = {128,128,128,128,128,128,  64,64,64,64,64,64};
constexpr int WW[12] = {128,128,128,150,150,150,  64,64,64,75,75,75};
constexpr int CI[12] = {0,0,1,0,1,2,  0,0,1,0,1,2};
constexpr int CJ[12] = {1,2,2,3,3,3,  1,2,2,3,3,3};
// Prefix offsets (floats) of each transposed plane (H,W,16) in workspace.
constexpr int OFF[12] = {0, 262144, 524288, 786432, 1093632, 1400832,
                         1708032, 1773568, 1839104, 1904640, 1981440, 2058240};

constexpr int NPTS       = 800000;            // B*N = 2*400000
constexpr int PTS_PER_WG = 64;
constexpr int ROW_STRIDE = 196;               // 192 dwords + 4 pad (16B aligned, 196%64==4 -> conflict-free b128)

// ---------------------------------------------------------------------------
// Pre-pass: (F,H,W) -> (H,W,F) so one texel's 16 features are 64 contiguous B.
// ---------------------------------------------------------------------------
__global__ __launch_bounds__(256)
void transpose_plane_kernel(const float* __restrict__ in, float* __restrict__ outp, int HW)
{
    int g = blockIdx.x * 256 + threadIdx.x;
    if (g < HW * 16) {
        int hw = g >> 4;
        int f  = g & 15;
        outp[g] = in[f * HW + hw];   // writes coalesced; reads form 16 strided 8B segments/wave
    }
}

// 4-weight bilinear blend: 1 mul + 3 fma per feature (weights hoisted per plane).
__device__ __forceinline__ float blend4(float v00, float v01, float v10, float v11,
                                        float w00, float w01, float w10, float w11)
{
    return fmaf(v11, w11, fmaf(v10, w10, fmaf(v01, w01, v00 * w00)));
}

// ---------------------------------------------------------------------------
// Main: lane = point (every VALU issue serves 32 points). Gather 4 corners
// (4x float4 each) from L2-resident transposed planes, blend, stage the
// 192-float output row in LDS, then stream the WG's contiguous 48KB chunk
// with async LDS->global b128 non-temporal stores (ASYNCcnt path).
// ---------------------------------------------------------------------------
__global__ __launch_bounds__(PTS_PER_WG)
void hexplane_kernel(const float* __restrict__ coords,
                     const float* __restrict__ planes,
                     float* __restrict__ out)
{
    extern __shared__ float smem[];   // PTS_PER_WG * ROW_STRIDE floats
    const int tid = threadIdx.x;
    const int p   = blockIdx.x * PTS_PER_WG + tid;      // grid is exact: p < NPTS always

    const float4 c4 = reinterpret_cast<const float4*>(coords)[p];
    const float cv[4] = {c4.x, c4.y, c4.z, c4.w};
    float* row = smem + tid * ROW_STRIDE;

#pragma unroll
    for (int k = 0; k < 12; ++k) {
        const int   W  = WW[k], H = HH[k];
        const float gx = cv[CI[k]];
        const float gy = cv[CJ[k]];

        float ix = (gx + 1.0f) * 0.5f * (float)(W - 1);
        ix = fminf(fmaxf(ix, 0.0f), (float)(W - 1));
        float iy = (gy + 1.0f) * 0.5f * (float)(H - 1);
        iy = fminf(fmaxf(iy, 0.0f), (float)(H - 1));

        const float x0f = floorf(ix), y0f = floorf(iy);
        const int   x0  = (int)x0f,   y0  = (int)y0f;
        const int   x1  = min(x0 + 1, W - 1);
        const int   y1  = min(y0 + 1, H - 1);
        const float wx  = ix - x0f,   wy  = iy - y0f;
        const float omx = 1.0f - wx,  omy = 1.0f - wy;
        const float w00 = omx * omy, w01 = wx * omy;
        const float w10 = omx * wy,  w11 = wx * wy;

        const float* pb = planes + OFF[k];
        const float4* r00 = reinterpret_cast<const float4*>(pb + (size_t)(y0 * W + x0) * 16);
        const float4* r01 = reinterpret_cast<const float4*>(pb + (size_t)(y0 * W + x1) * 16);
        const float4* r10 = reinterpret_cast<const float4*>(pb + (size_t)(y1 * W + x0) * 16);
        const float4* r11 = reinterpret_cast<const float4*>(pb + (size_t)(y1 * W + x1) * 16);

#pragma unroll
        for (int cc = 0; cc < 4; ++cc) {
            const float4 a = r00[cc], b = r01[cc], g = r10[cc], d = r11[cc];
            float4 res;
            res.x = blend4(a.x, b.x, g.x, d.x, w00, w01, w10, w11);
            res.y = blend4(a.y, b.y, g.y, d.y, w00, w01, w10, w11);
            res.z = blend4(a.z, b.z, g.z, d.z, w00, w01, w10, w11);
            res.w = blend4(a.w, b.w, g.w, d.w, w00, w01, w10, w11);
            *reinterpret_cast<float4*>(row + k * 16 + cc * 4) = res;   // ds_store_b128
        }
    }

    __syncthreads();

    // Async LDS -> global streaming of this WG's contiguous output chunk.
    // 64 rows x 48 float4 = 3072 chunks; 48 per thread; ASYNCcnt <= 48 < 63.
    // Non-temporal: output is write-once (614 MB) -- keep L2 for the planes.
    const uint64_t sbase =
        (uint64_t)(uintptr_t)(out + (size_t)blockIdx.x * (PTS_PER_WG * 192));
    for (int e = tid; e < PTS_PER_WG * 48; e += PTS_PER_WG) {
        const int r  = e / 48;
        const int cc = e - r * 48;
        const unsigned goff = (unsigned)((r * 192 + cc * 4) * 4);                 // byte offset, 16B aligned
        const unsigned lds  = (unsigned)(uintptr_t)(&smem[r * ROW_STRIDE + cc * 4]); // low 32 bits = LDS byte offset
        asm volatile("global_store_async_from_lds_b128 %0, %1, %2 th:TH_STORE_NT"
                     :: "v"(goff), "v"(lds), "s"(sbase) : "memory");
    }
    asm volatile("s_wait_asynccnt 0x0" ::: "memory");
}

// ---------------------------------------------------------------------------
extern "C" void kernel_launch(void* const* d_in, const int* in_sizes, int n_in,
                              void* d_out, int out_size, void* d_ws, size_t ws_size,
                              hipStream_t stream)
{
    (void)in_sizes; (void)n_in; (void)out_size; (void)ws_size;
    const float* coords = (const float*)d_in[0];
    float*       ws     = (float*)d_ws;   // needs 8.54 MB

    // Transpose the 12 planes into workspace (channels-last).
    for (int k = 0; k < 12; ++k) {
        const int HW = HH[k] * WW[k];
        const int n  = HW * 16;
        const int grid = (n + 255) / 256;
        transpose_plane_kernel<<<grid, 256, 0, stream>>>(
            (const float*)d_in[1 + k], ws + OFF[k], HW);
    }

    // Main sampling kernel: exact grid, 50 KB dynamic LDS per WG.
    const int nblocks = NPTS / PTS_PER_WG;                 // 12500
    const size_t lds_bytes = PTS_PER_WG * ROW_STRIDE * sizeof(float); // 50176
    hexplane_kernel<<<nblocks, PTS_PER_WG, lds_bytes, stream>>>(
        coords, ws, (float*)d_out);
}